// VectorQuantizer2_28295244546658
// MI455X (gfx1250) — compile-verified
//
#include <hip/hip_runtime.h>
#include <hip/hip_bf16.h>
#include <stdint.h>

// Problem constants (from the reference)
#define M_TOTAL 8192     // 8 * 1024 latent vectors
#define NE      16384    // codebook entries
#define D_DIM   256      // embedding dim
#define BETA    0.25f

#define BM      32       // z rows per workgroup (2 WMMA M-subtiles per wave)

typedef __attribute__((ext_vector_type(2))) float v2f;
typedef __attribute__((ext_vector_type(8))) float v8f;

#define LDS_PITCH 260    // zs pitch: rows land 4 banks apart -> conflict-free A frags
#define EPITCH    68     // B staging pitch (floats): 68%64==4 -> conflict-free B frags
#define QK        64     // K elements staged per phase (quarter of D_DIM)
#define NPHASES   (NE / 128 * 4)   // 128 tiles x 4 quarter-K phases = 512

// One workgroup = 32 z rows in LDS; 8 waves each own a 16-wide N subtile and
// sweep N=16384 in 128-tile chunks. Codebook tiles are staged into wave-private
// LDS buffers with GLOBAL_LOAD_ASYNC_TO_LDS_B128 (ASYNCcnt, double-buffered
// quarter-K phases overlapping the WMMAs). Each staged B fragment feeds two
// v_wmma_f32_16x16x4_f32 ops. Fused ||e||^2 + argmin + STE/loss epilogue.
__global__ __launch_bounds__(256) void vq_argmin_wmma_kernel(
    const float* __restrict__ z,
    const float* __restrict__ cb,
    float* __restrict__ out) {
  __shared__ float zs[BM * LDS_PITCH];
  __shared__ float ebuf[8 * 2 * 16 * EPITCH];   // [wave][buf][row][col]
  __shared__ float wmin[8][BM];
  __shared__ int   widx[8][BM];
  __shared__ int   rowidx[BM];

  const int tid  = threadIdx.x;
  const int wave = tid >> 5;
  const int lane = tid & 31;
  const int m0   = blockIdx.x * BM;

  // Stage the 32x256 Z block into LDS (coalesced: thread t = column t).
  #pragma unroll
  for (int j = 0; j < BM; ++j) {
    zs[j * LDS_PITCH + tid] = z[(size_t)(m0 + j) * D_DIM + tid];
  }
  __syncthreads();

  const int lrow = lane & 15;        // row within A subtile / col within B tile
  const int koff = (lane >> 4) * 2;  // K phase: lanes 0-15 -> K,K+1; 16-31 -> K+2,K+3

  // ---- async staging addresses -------------------------------------------
  // Batch of 8 x b128: instr j moves the quarter-K chunks of rows 2j,2j+1.
  //   lane l: global = cb + (n0 + (l>>4))*1024 + (l&15)*16  (+ q*256 + j*2048)
  //           lds    = wavebuf + (2j + (l>>4))*272 + (l&15)*16
  const uint64_t gl0 =
      (uint64_t)(uintptr_t)cb + (uint64_t)(wave * 16 + (lane >> 4)) * 1024u +
      (uint64_t)(lane & 15) * 16u;
  const unsigned ebase =
      (unsigned)(uintptr_t)(void*)ebuf + (unsigned)wave * (2u * 16u * EPITCH * 4u) +
      (unsigned)(lane >> 4) * (EPITCH * 4u) + (unsigned)(lane & 15) * 16u;
  const unsigned BUFB = 16u * EPITCH * 4u;   // 4352B: one 16x64 staged quarter

  // Issue one phase batch (8 async b128 gathers -> wave-private LDS buffer).
  auto issue = [&](unsigned lbuf, uint64_t g) {
    // Close WAR on the buffer: prior ds reads must land before DMA overwrites.
    asm volatile("s_wait_dscnt 0x0" ::: "memory");
    #pragma unroll
    for (int j = 0; j < 8; ++j) {
      asm volatile("global_load_async_to_lds_b128 %0, %1, off"
                   :: "v"(lbuf), "v"(g) : "memory");
      lbuf += 2u * EPITCH * 4u;   // two LDS rows
      g    += 2048u;              // two global rows
    }
  };

  float minv0[8], minv1[8];
  int   mini0[8], mini1[8];
  #pragma unroll
  for (int r = 0; r < 8; ++r) {
    minv0[r] = 3.4e38f; mini0[r] = 0;
    minv1[r] = 3.4e38f; mini1[r] = 0;
  }

  const float* a0row = &zs[lrow * LDS_PITCH + koff];          // rows 0..15
  const float* a1row = &zs[(16 + lrow) * LDS_PITCH + koff];   // rows 16..31

  // Software pipeline: phase p+1 DMA overlaps phase p compute.
  uint64_t gph    = gl0;       // current phase global base
  unsigned bufsel = 0;
  int      phase  = 0;

  issue(ebase, gph);           // prologue: phase 0 -> buf 0

  const int ITERS = NE / 128;
  for (int it = 0; it < ITERS; ++it) {
    v8f   acc0  = {};          // rows r / 8+r of the block
    v8f   acc1  = {};          // rows 16+r / 24+r
    float epart = 0.0f;        // partial ||e||^2 for row n0+lrow (this K slice)

    for (int q = 0; q < 4; ++q) {
      const uint64_t gnext = gph + ((q == 3) ? (uint64_t)(128 * 1024 - 3 * 256)
                                             : (uint64_t)256);
      if (phase + 1 < NPHASES) {
        issue(ebase + (bufsel ^ 1) * BUFB, gnext);
        asm volatile("s_wait_asynccnt 0x8" ::: "memory");  // oldest batch done
      } else {
        asm volatile("s_wait_asynccnt 0x0" ::: "memory");  // final batch done
      }

      const float* ap0 = a0row + q * QK;
      const float* ap1 = a1row + q * QK;
      const float* bp  = &ebuf[wave * (2 * 16 * EPITCH) + bufsel * (16 * EPITCH) +
                               lrow * EPITCH + koff];

      #pragma unroll 8
      for (int kk = 0; kk < QK; kk += 4) {
        v2f b  = *(const v2f*)(bp + kk);    // B: 4x16 f32 fragment from LDS
        v2f a0 = *(const v2f*)(ap0 + kk);   // A: 16x4 f32 fragments from LDS
        v2f a1 = *(const v2f*)(ap1 + kk);
        epart = fmaf(b.x, b.x, epart);
        epart = fmaf(b.y, b.y, epart);
        acc0 = __builtin_amdgcn_wmma_f32_16x16x4_f32(
            false, a0, false, b, (short)0, acc0, false, false);
        acc1 = __builtin_amdgcn_wmma_f32_16x16x4_f32(
            false, a1, false, b, (short)0, acc1, false, false);
      }

      gph = gnext;
      bufsel ^= 1;
      ++phase;
    }

    // Complete ||e||^2: combine the two K-half partials (lanes L and L^16
    // both hold column n0+lrow of the score tile).
    const float enorm = epart + __shfl_xor(epart, 16, 32);
    const int   ncol  = it * 128 + wave * 16 + lrow;

    // score = ||e||^2 - 2*(z.e); ||z||^2 is row-constant -> argmin unchanged.
    #pragma unroll
    for (int r = 0; r < 8; ++r) {
      float s0 = fmaf(-2.0f, acc0[r], enorm);
      float s1 = fmaf(-2.0f, acc1[r], enorm);
      if (s0 < minv0[r]) { minv0[r] = s0; mini0[r] = ncol; }  // n increases -> first-min kept
      if (s1 < minv1[r]) { minv1[r] = s1; mini1[r] = ncol; }
    }
  }

  // Butterfly min-reduce across the 16 lanes of each half-wave
  // (acc0: lanes 0-15 -> row r, lanes 16-31 -> row 8+r; acc1: +16).
  #pragma unroll
  for (int r = 0; r < 8; ++r) {
    float v0 = minv0[r]; int i0 = mini0[r];
    float v1 = minv1[r]; int i1 = mini1[r];
    #pragma unroll
    for (int m = 1; m < 16; m <<= 1) {
      float ov0 = __shfl_xor(v0, m, 32); int oi0 = __shfl_xor(i0, m, 32);
      float ov1 = __shfl_xor(v1, m, 32); int oi1 = __shfl_xor(i1, m, 32);
      if (ov0 < v0 || (ov0 == v0 && oi0 < i0)) { v0 = ov0; i0 = oi0; }
      if (ov1 < v1 || (ov1 == v1 && oi1 < i1)) { v1 = ov1; i1 = oi1; }
    }
    minv0[r] = v0; mini0[r] = i0;
    minv1[r] = v1; mini1[r] = i1;
  }
  if (lane == 0) {
    #pragma unroll
    for (int r = 0; r < 8; ++r) {
      wmin[wave][r]      = minv0[r]; widx[wave][r]      = mini0[r];
      wmin[wave][16 + r] = minv1[r]; widx[wave][16 + r] = mini1[r];
    }
  } else if (lane == 16) {
    #pragma unroll
    for (int r = 0; r < 8; ++r) {
      wmin[wave][8 + r]  = minv0[r]; widx[wave][8 + r]  = mini0[r];
      wmin[wave][24 + r] = minv1[r]; widx[wave][24 + r] = mini1[r];
    }
  }
  __syncthreads();

  // Cross-wave reduce: 8 candidates per row (ties -> smaller index,
  // matching jnp.argmin first-occurrence semantics).
  if (tid < BM) {
    float bv = wmin[0][tid]; int bi = widx[0][tid];
    #pragma unroll
    for (int w = 1; w < 8; ++w) {
      float v = wmin[w][tid]; int i = widx[w][tid];
      if (v < bv || (v == bv && i < bi)) { bv = v; bi = i; }
    }
    rowidx[tid] = bi;
  }
  __syncthreads();

  // Epilogue: outputs concatenated as (z_q_st[f32 M*D], indices[i32 M], loss[f32 M*D]).
  float* zq_out   = out;
  int*   idx_out  = (int*)(out + (size_t)M_TOTAL * D_DIM);
  float* loss_out = out + (size_t)M_TOTAL * D_DIM + M_TOTAL;

  if (tid < BM) idx_out[m0 + tid] = rowidx[tid];

  #pragma unroll
  for (int rr = 0; rr < BM; ++rr) {
    const int   nidx = rowidx[rr];
    const float e    = cb[(size_t)nidx * D_DIM + tid];  // coalesced row read
    const float zz   = zs[rr * LDS_PITCH + tid];
    const float dlt  = e - zz;
    zq_out[(size_t)(m0 + rr) * D_DIM + tid]   = e;                      // STE value == z_q
    loss_out[(size_t)(m0 + rr) * D_DIM + tid] = (1.0f + BETA) * dlt * dlt;
  }
}

extern "C" void kernel_launch(void* const* d_in, const int* in_sizes, int n_in,
                              void* d_out, int out_size, void* d_ws, size_t ws_size,
                              hipStream_t stream) {
  (void)in_sizes; (void)n_in; (void)out_size; (void)d_ws; (void)ws_size;
  const float* z  = (const float*)d_in[0];
  const float* cb = (const float*)d_in[1];
  float* out = (float*)d_out;

  dim3 grid(M_TOTAL / BM);   // 256 workgroups
  dim3 block(256);           // 8 waves (wave32)
  vq_argmin_wmma_kernel<<<grid, block, 0, stream>>>(z, cb, out);
}